// AdditiveAttention_80753975099802
// MI455X (gfx1250) — compile-verified
//
#include <hip/hip_runtime.h>
#include <hip/hip_bf16.h>
#include <stdint.h>

#define MASK_VALUE -1000000.0f

typedef __attribute__((ext_vector_type(2))) float v2f;
typedef __attribute__((ext_vector_type(8))) float v8f;

// Hardware tanh (gfx1250 V_TANH_F32) with safe fallback.
__device__ __forceinline__ float hw_tanhf(float x) {
#if defined(__has_builtin)
#if __has_builtin(__builtin_amdgcn_tanhf)
    return __builtin_amdgcn_tanhf(x);
#else
    return tanhf(x);
#endif
#else
    return tanhf(x);
#endif
}

// ---------------------------------------------------------------------------
// Batched fp32 GEMM via V_WMMA_F32_16X16X4_F32 (wave32).
// One wave computes a 16(M) x 64(N) tile of C = A(MxK) * B(KxN), both
// row-major, so each A fragment is reused by 4 WMMAs.
// Fragment layouts per CDNA5 ISA 7.12.2:
//   A 16x4 : lanes 0-15 -> row M=lane,    v0=A[M,k0+0], v1=A[M,k0+1]
//            lanes16-31 -> row M=lane-16, v0=A[M,k0+2], v1=A[M,k0+3]
//   B 4x16 : lanes 0-15 -> col N=lane,    v0=B[k0+0,N], v1=B[k0+1,N]
//            lanes16-31 -> col N=lane-16, v0=B[k0+2,N], v1=B[k0+3,N]
//   C/D    : VGPR i: lanes 0-15 -> M=i, lanes 16-31 -> M=i+8; N=lane%16
// ---------------------------------------------------------------------------
__global__ __launch_bounds__(32)
void gemm_wmma_f32_16x64(const float* __restrict__ A, int lda, int64_t strideA,
                         const float* __restrict__ B, int ldb, int64_t strideB,
                         float* __restrict__ C, int ldc, int64_t strideC,
                         int K) {
    const int batch = blockIdx.z;
    const int tm    = blockIdx.y;   // 16-row tile of C
    const int tn    = blockIdx.x;   // 64-col tile of C

    const float* Ab = A + batch * strideA + (int64_t)tm * 16 * lda;
    const float* Bb = B + batch * strideB + tn * 64;
    float*       Cb = C + batch * strideC + (int64_t)tm * 16 * ldc + tn * 64;

    const int lane = threadIdx.x;        // 0..31 (wave32)
    const int half = lane >> 4;          // 0 or 1
    const int r    = lane & 15;          // row (A) / col (B,C)

    v8f acc0 = {}, acc1 = {}, acc2 = {}, acc3 = {};

    const float* arow = Ab + (int64_t)r * lda;

    for (int k0 = 0; k0 < K; k0 += 4) {
        // A fragment: two consecutive K values starting at k0 + 2*half
        v2f a = *(const v2f*)(arow + k0 + 2 * half);

        // B fragments for 4 adjacent 16-wide N tiles:
        // rows k0+2*half and k0+2*half+1, columns r, r+16, r+32, r+48
        const float* bp = Bb + (int64_t)(k0 + 2 * half) * ldb + r;
        v2f b0, b1, b2, b3;
        b0.x = bp[0];   b0.y = bp[ldb];
        b1.x = bp[16];  b1.y = bp[16 + ldb];
        b2.x = bp[32];  b2.y = bp[32 + ldb];
        b3.x = bp[48];  b3.y = bp[48 + ldb];

        // Prefetch two K-steps ahead (global_prefetch_b8; L2-resident data,
        // hides the short L2 latency in this latency-bound problem).
        if (k0 + 8 < K) {
            __builtin_prefetch(arow + k0 + 8, 0, 1);
            __builtin_prefetch(Bb + (int64_t)(k0 + 8 + 2 * half) * ldb + r, 0, 1);
        }

        acc0 = __builtin_amdgcn_wmma_f32_16x16x4_f32(false, a, false, b0,
                                                     (short)0, acc0, false, false);
        acc1 = __builtin_amdgcn_wmma_f32_16x16x4_f32(false, a, false, b1,
                                                     (short)0, acc1, false, false);
        acc2 = __builtin_amdgcn_wmma_f32_16x16x4_f32(false, a, false, b2,
                                                     (short)0, acc2, false, false);
        acc3 = __builtin_amdgcn_wmma_f32_16x16x4_f32(false, a, false, b3,
                                                     (short)0, acc3, false, false);
    }

#pragma unroll
    for (int i = 0; i < 8; ++i) {
        float* crow = Cb + (int64_t)(i + 8 * half) * ldc + r;
        crow[0]  = acc0[i];
        crow[16] = acc1[i];
        crow[32] = acc2[i];
        crow[48] = acc3[i];
    }
}

// ---------------------------------------------------------------------------
// Fused additive scores + mask + softmax.
// grid = (Lq, B), block = 256 (8 waves). One workgroup per (b, q) row.
//   score[k] = sum_h wv[h] * tanh(qh[b,q,h] + kh[b,k,h])   (HW V_TANH_F32)
// then masked softmax over k (mask: k >= valid_len -> -1e6, as in reference).
// ---------------------------------------------------------------------------
__global__ __launch_bounds__(256)
void scores_softmax_kernel(const float* __restrict__ qh,   // (B*Lq, 256)
                           const float* __restrict__ kh,   // (B*Lk, 256)
                           const float* __restrict__ wv,   // (256,)
                           const int*   __restrict__ valid_lens, // (B,)
                           float*       __restrict__ attn) // (B*Lq, 512)
{
    constexpr int H  = 256;
    constexpr int LK = 512;
    constexpr int LQ = 128;

    const int q = blockIdx.x;
    const int b = blockIdx.y;
    const int tid  = threadIdx.x;
    const int lane = tid & 31;
    const int wave = tid >> 5;           // 0..7

    __shared__ float s_q[H];
    __shared__ float s_wv[H];
    __shared__ float s_sc[LK];
    __shared__ float s_red[8];

    s_q[tid]  = qh[((int64_t)(b * LQ + q)) * H + tid];
    s_wv[tid] = wv[tid];
    __syncthreads();

    // Each wave owns k-columns wave, wave+8, ... ; lanes split H coalesced.
    for (int k = wave; k < LK; k += 8) {
        const float* krow = kh + ((int64_t)(b * LK + k)) * H;
        float acc = 0.0f;
#pragma unroll
        for (int j = 0; j < 8; ++j) {
            const int h = lane + 32 * j;
            acc += s_wv[h] * hw_tanhf(s_q[h] + krow[h]);
        }
        // wave32 reduction
#pragma unroll
        for (int off = 16; off > 0; off >>= 1)
            acc += __shfl_xor(acc, off, 32);
        if (lane == 0) s_sc[k] = acc;
    }
    __syncthreads();

    // Masked softmax over 512 entries; 256 threads own 2 entries each.
    const int vl = valid_lens[b];
    float v0 = (tid       >= vl) ? MASK_VALUE : s_sc[tid];
    float v1 = (tid + 256 >= vl) ? MASK_VALUE : s_sc[tid + 256];

    // --- block max ---
    float m = fmaxf(v0, v1);
#pragma unroll
    for (int off = 16; off > 0; off >>= 1)
        m = fmaxf(m, __shfl_xor(m, off, 32));
    if (lane == 0) s_red[wave] = m;
    __syncthreads();
    if (tid == 0) {
        float t = s_red[0];
#pragma unroll
        for (int i = 1; i < 8; ++i) t = fmaxf(t, s_red[i]);
        s_red[0] = t;
    }
    __syncthreads();
    m = s_red[0];
    __syncthreads();

    // --- exp + block sum (v_exp_f32 path) ---
    const float e0 = __expf(v0 - m);
    const float e1 = __expf(v1 - m);
    float s = e0 + e1;
#pragma unroll
    for (int off = 16; off > 0; off >>= 1)
        s += __shfl_xor(s, off, 32);
    if (lane == 0) s_red[wave] = s;
    __syncthreads();
    if (tid == 0) {
        float t = 0.0f;
#pragma unroll
        for (int i = 0; i < 8; ++i) t += s_red[i];
        s_red[0] = t;
    }
    __syncthreads();
    const float inv = 1.0f / s_red[0];

    float* arow = attn + ((int64_t)(b * LQ + q)) * LK;
    arow[tid]       = e0 * inv;
    arow[tid + 256] = e1 * inv;
}

// ---------------------------------------------------------------------------
// Launcher. Inputs (setup_inputs order):
//   0 queries (8,128,256) f32   1 keys (8,512,256) f32   2 values (8,512,256) f32
//   3 valid_lens (8,) i32       4 Wq (256,256) f32       5 Wk (256,256) f32
//   6 wv (256,) f32
// Output: (8,128,256) f32
// ---------------------------------------------------------------------------
extern "C" void kernel_launch(void* const* d_in, const int* in_sizes, int n_in,
                              void* d_out, int out_size, void* d_ws, size_t ws_size,
                              hipStream_t stream) {
    const float* dq  = (const float*)d_in[0];
    const float* dk  = (const float*)d_in[1];
    const float* dv  = (const float*)d_in[2];
    const int*   vln = (const int*)d_in[3];
    const float* Wq  = (const float*)d_in[4];
    const float* Wk  = (const float*)d_in[5];
    const float* wv  = (const float*)d_in[6];
    float*       out = (float*)d_out;

    constexpr int B = 8, LQ = 128, LK = 512, H = 256, DV = 256;

    // Workspace layout: qh (1 MB) | kh (4 MB) | attn (2 MB)
    char* ws = (char*)d_ws;
    float* qh   = (float*)ws;
    float* kh   = (float*)(ws + (size_t)B * LQ * H * sizeof(float));
    float* attn = (float*)(ws + (size_t)B * LQ * H * sizeof(float)
                              + (size_t)B * LK * H * sizeof(float));

    // qh = Q_flat (1024x256) @ Wq (256x256)
    gemm_wmma_f32_16x64<<<dim3(H / 64, (B * LQ) / 16, 1), 32, 0, stream>>>(
        dq, H, 0, Wq, H, 0, qh, H, 0, /*K=*/H);

    // kh = K_flat (4096x256) @ Wk (256x256)
    gemm_wmma_f32_16x64<<<dim3(H / 64, (B * LK) / 16, 1), 32, 0, stream>>>(
        dk, H, 0, Wk, H, 0, kh, H, 0, /*K=*/H);

    // attn = masked_softmax( tanh(qh + kh) . wv )
    scores_softmax_kernel<<<dim3(LQ, B), 256, 0, stream>>>(qh, kh, wv, vln, attn);

    // out[b] = attn[b] (128x512) @ V[b] (512x256)   (batched)
    gemm_wmma_f32_16x64<<<dim3(DV / 64, LQ / 16, B), 32, 0, stream>>>(
        attn, LK, (int64_t)LQ * LK,
        dv,   DV, (int64_t)LK * DV,
        out,  DV, (int64_t)LQ * DV,
        /*K=*/LK);
}